// HierEncoder_32435593019950
// MI455X (gfx1250) — compile-verified
//
#include <hip/hip_runtime.h>

typedef __attribute__((ext_vector_type(2))) float v2f;
typedef __attribute__((ext_vector_type(8))) float v8f;
typedef __attribute__((ext_vector_type(4))) unsigned int u32x4;
typedef __attribute__((ext_vector_type(8))) int i32x8;
typedef __attribute__((ext_vector_type(4))) int i32x4;

#define N_SOURCE   200000
#define N_TARGET   20000
#define SOURCE_DIM 1024
#define EMBED_DIM  32
#define TARGET_DIM 16
#define N_EDGES    4000000

// B staging geometry: 8 chunks of 128 K-rows x 32 cols of embed.
// TDM pads LDS every 32 DWORDs by 8 DWORDs -> row stride 40 floats.
// Stride 40: lane-half row delta (2 rows = 80 floats) == 16 mod 64 banks,
// so the two 16-lane halves of each ds_load_2addr_b32 hit disjoint banks.
#define CHUNK_K     128
#define N_CHUNKS    (SOURCE_DIM / CHUNK_K)          // 8
#define ROW_STRIDE  40                              // floats per padded LDS row
#define CHUNK_LDSF  (CHUNK_K * ROW_STRIDE)          // 5120 floats = 20480 B

// ---------------------------------------------------------------------------
// TDM: DMA one embed chunk (128 x 32 fp32, contiguous 16 KB) into LDS with
// row padding. D# layout per CDNA5 ISA ch.8 (group0 128b, group1 256b).
// Issued by a single wave; completion via TENSORcnt.
// This toolchain's builtin takes 6 args: (g0, g1, g2, g3, g_extra, cpol).
// ---------------------------------------------------------------------------
#if __has_builtin(__builtin_amdgcn_tensor_load_to_lds)
#define USE_TDM 1
__device__ __forceinline__ void tdm_load_chunk(const float* gsrc, unsigned lds_off) {
  unsigned long long ga = (unsigned long long)(__SIZE_TYPE__)gsrc;
  u32x4 g0;
  g0[0] = 1u;                                            // count=1, user D#
  g0[1] = lds_off;                                       // LDS byte address
  g0[2] = (unsigned)(ga & 0xFFFFFFFFu);                  // global_addr[31:0]
  g0[3] = (unsigned)((ga >> 32) & 0x1FFFFFFu)            // global_addr[56:32]
        | (2u << 30);                                    // type=2 (image)
  i32x8 g1;
  g1[0] = (2 << 16)                                      // data_size = 4 B
        | (1 << 20)                                      // pad_enable
        | (4 << 22)                                      // pad_interval: 32 DWORDs
        | (7 << 25);                                     // pad_amount: 8 DWORDs
  g1[1] = (int)((unsigned)(CHUNK_K * EMBED_DIM) << 16);  // tensor_dim0 lo16
  g1[2] = (int)(1u << 16);                               // dim0 hi=0 | tensor_dim1=1
  g1[3] = (int)((unsigned)(CHUNK_K * EMBED_DIM) << 16);  // tile_dim0 = 4096
  g1[4] = 0;                                             // tile_dim1/2 unused
  g1[5] = CHUNK_K * EMBED_DIM;                           // tensor_dim0_stride lo
  g1[6] = 0;
  g1[7] = 0;
  i32x4 z4 = {0, 0, 0, 0};                               // groups 2/3: unused (<=2D)
  i32x8 z8 = {0, 0, 0, 0, 0, 0, 0, 0};
  __builtin_amdgcn_tensor_load_to_lds(g0, g1, z4, z4, z8, 0);
}
#else
#define USE_TDM 0
#endif

// ---------------------------------------------------------------------------
// Kernel 1: x = (source_feat @ embed) / x_norm[:,None] -> ws_x [N_SOURCE][32]
// One wave32 per 16-row tile, 8 waves/block. embed chunks staged into LDS by
// the Tensor Data Mover, double-buffered so the DMA hides behind the 64 WMMAs
// per chunk. Inner loop: 1 global b64 (A) + 2 ds_load_2addr_b32 (B) + 2 WMMA.
// ---------------------------------------------------------------------------
__launch_bounds__(256)
__global__ void k_embed_gemm(const float* __restrict__ feat,
                             const float* __restrict__ xnorm,
                             const float* __restrict__ embed,
                             float* __restrict__ x_out) {
  __shared__ float sB[2][CHUNK_LDSF];        // 2 x 20 KB
  const int  tid  = threadIdx.x;
  const int  lane = tid & 31;
  const int  wave = tid >> 5;
  const int  half = lane >> 4;               // lane-half -> K offset {0,2}
  const int  lr   = lane & 15;
  const int  koff = 2 * half;
  const int  tile = blockIdx.x * 8 + wave;
  const bool valid = (tile * 16) < N_SOURCE;
  const long m0 = (long)tile * 16;

  v8f acc0 = {};                             // N = 0..15
  v8f acc1 = {};                             // N = 16..31
  const float* arow = feat + (m0 + lr) * SOURCE_DIM + koff;

#if USE_TDM
  const unsigned ldsBase = (unsigned)(unsigned long long)(__SIZE_TYPE__)&sB[0][0];
  if (wave == 0) {
    tdm_load_chunk(embed, ldsBase);
    __builtin_amdgcn_s_wait_tensorcnt(0);
  }
  __syncthreads();

  for (int c = 0; c < N_CHUNKS; ++c) {
    const int buf = c & 1;
    if (wave == 0 && (c + 1) < N_CHUNKS)     // prefetch next chunk via TDM
      tdm_load_chunk(embed + (long)(c + 1) * CHUNK_K * EMBED_DIM,
                     ldsBase + (unsigned)((buf ^ 1) * sizeof(float) * CHUNK_LDSF));
    if (valid) {                             // wave-uniform: EXEC all-1s for WMMA
      const float* bbase = &sB[buf][0];
#pragma unroll 4
      for (int s = 0; s < CHUNK_K / 4; ++s) {
        v2f a = *(const v2f*)(arow + c * CHUNK_K + 4 * s);
        const float* bp = bbase + (4 * s + koff) * ROW_STRIDE + lr;
        v2f b0, b1;
        b0.x = bp[0];              b0.y = bp[ROW_STRIDE];
        b1.x = bp[16];             b1.y = bp[ROW_STRIDE + 16];
        acc0 = __builtin_amdgcn_wmma_f32_16x16x4_f32(false, a, false, b0,
                                                     (short)0, acc0, false, false);
        acc1 = __builtin_amdgcn_wmma_f32_16x16x4_f32(false, a, false, b1,
                                                     (short)0, acc1, false, false);
      }
    }
    __syncthreads();                         // everyone done reading buf
    if (wave == 0) __builtin_amdgcn_s_wait_tensorcnt(0);
    __syncthreads();                         // next chunk's LDS data visible
  }
#else
  // Fallback (no TDM builtin): cooperative VALU staging, single-buffered.
  for (int c = 0; c < N_CHUNKS; ++c) {
    __syncthreads();
    for (int i = tid; i < CHUNK_K * EMBED_DIM; i += 256) {
      const int r = i >> 5, col = i & 31;
      sB[0][r * ROW_STRIDE + col] = embed[(long)(c * CHUNK_K + r) * EMBED_DIM + col];
    }
    __syncthreads();
    if (valid) {
#pragma unroll 4
      for (int s = 0; s < CHUNK_K / 4; ++s) {
        v2f a = *(const v2f*)(arow + c * CHUNK_K + 4 * s);
        const float* bp = &sB[0][(4 * s + koff) * ROW_STRIDE + lr];
        v2f b0, b1;
        b0.x = bp[0];              b0.y = bp[ROW_STRIDE];
        b1.x = bp[16];             b1.y = bp[ROW_STRIDE + 16];
        acc0 = __builtin_amdgcn_wmma_f32_16x16x4_f32(false, a, false, b0,
                                                     (short)0, acc0, false, false);
        acc1 = __builtin_amdgcn_wmma_f32_16x16x4_f32(false, a, false, b1,
                                                     (short)0, acc1, false, false);
      }
    }
  }
#endif

  if (valid) {
    // D layout: VGPR v, lane l -> row M = v + 8*(l>=16), col N = l&15
#pragma unroll
    for (int v = 0; v < 8; ++v) {
      const long row = m0 + v + 8 * half;
      const float inv = 1.0f / xnorm[row];
      x_out[row * EMBED_DIM + lr]      = acc0[v] * inv;
      x_out[row * EMBED_DIM + 16 + lr] = acc1[v] * inv;
    }
  }
}

// ---------------------------------------------------------------------------
// Kernel 2: per-target mean over edge slice. One wave32 per target; lane =
// embedding dim so every gather x[src] is one coalesced 128B line (the 25.6 MB
// x table is L2-resident). Index stream prefetched; batches broadcast by shfl.
// ---------------------------------------------------------------------------
__launch_bounds__(256)
__global__ void k_aggregate(const float* __restrict__ x,
                            const int* __restrict__ edge_src,
                            const int* __restrict__ range_list,
                            float* __restrict__ agg) {
  const int wave = (blockIdx.x * blockDim.x + threadIdx.x) >> 5;
  const int lane = threadIdx.x & 31;
  if (wave >= N_TARGET) return;

  const int start = range_list[2 * wave];
  const int end   = range_list[2 * wave + 1];

  float acc = 0.0f;
  for (int e0 = start; e0 < end; e0 += 32) {
    __builtin_prefetch(edge_src + e0 + 64, 0, 0);   // global_prefetch_b8
    const int myE   = e0 + lane;
    const int mySrc = (myE < end) ? edge_src[myE] : 0;
    const int cnt   = min(32, end - e0);
    for (int j = 0; j < cnt; ++j) {
      const int s = __shfl(mySrc, j, 32);
      acc += x[(long)s * EMBED_DIM + lane];
    }
  }
  const float deg = (float)(end - start);
  agg[(long)wave * EMBED_DIM + lane] = acc * (1.0f / fmaxf(deg, 1.0f));
}

// ---------------------------------------------------------------------------
// Kernel 3: out = agg @ weight  ([20000x32]x[32x16]) via f32 WMMA.
// ---------------------------------------------------------------------------
__launch_bounds__(256)
__global__ void k_proj_gemm(const float* __restrict__ agg,
                            const float* __restrict__ weight,
                            float* __restrict__ out) {
  const int lane = threadIdx.x & 31;
  const int wave = threadIdx.x >> 5;
  const int half = lane >> 4;
  const int lr   = lane & 15;
  const int tile = blockIdx.x * 8 + wave;
  if (tile * 16 >= N_TARGET) return;
  const long m0 = (long)tile * 16;

  v8f acc = {};
#pragma unroll
  for (int s = 0; s < EMBED_DIM / 4; ++s) {
    const int k0 = s * 4 + 2 * half;
    v2f a = *(const v2f*)(agg + (m0 + lr) * EMBED_DIM + k0);
    v2f b;
    b.x = weight[(long)k0 * TARGET_DIM + lr];
    b.y = weight[(long)(k0 + 1) * TARGET_DIM + lr];
    acc = __builtin_amdgcn_wmma_f32_16x16x4_f32(false, a, false, b,
                                                (short)0, acc, false, false);
  }
#pragma unroll
  for (int v = 0; v < 8; ++v) {
    const long row = m0 + v + 8 * half;
    out[row * TARGET_DIM + lr] = acc[v];
  }
}

// ---------------------------------------------------------------------------
extern "C" void kernel_launch(void* const* d_in, const int* in_sizes, int n_in,
                              void* d_out, int out_size, void* d_ws, size_t ws_size,
                              hipStream_t stream) {
  const float* feat   = (const float*)d_in[0];   // [200000,1024] f32
  const float* xnorm  = (const float*)d_in[1];   // [200000] f32
  const int*   edge   = (const int*)d_in[2];     // [2,4000000] i32 (row0 = src)
  const int*   rl     = (const int*)d_in[3];     // [20000,2] i32
  const float* embed  = (const float*)d_in[4];   // [1024,32] f32
  const float* weight = (const float*)d_in[5];   // [32,16] f32
  float*       out    = (float*)d_out;           // [20000,16] f32

  float* x_ws = (float*)d_ws;                            // 200000*32 f32 = 25.6 MB
  float* agg  = x_ws + (size_t)N_SOURCE * EMBED_DIM;     //  20000*32 f32 =  2.6 MB

  const int mtiles1 = N_SOURCE / 16;   // 12500
  const int mtiles3 = N_TARGET / 16;   // 1250

  k_embed_gemm<<<(mtiles1 + 7) / 8, 256, 0, stream>>>(feat, xnorm, embed, x_ws);
  k_aggregate<<<(N_TARGET * 32 + 255) / 256, 256, 0, stream>>>(x_ws, edge, rl, agg);
  k_proj_gemm<<<(mtiles3 + 7) / 8, 256, 0, stream>>>(agg, weight, out);
}